// ExperimentalNetwork_66915590471785
// MI455X (gfx1250) — compile-verified
//
#include <hip/hip_runtime.h>

#define B_ 1024
#define L_ 512
#define E_ 300
#define H_ 512
#define O_ 5

typedef __attribute__((ext_vector_type(2))) float v2f;
typedef __attribute__((ext_vector_type(8))) float v8f;

// ---------------------------------------------------------------------------
// Kernel A: per-row token staging + first-zero length + gather-sum + mean.
// One block per batch row b. Tokens -> LDS, then 300 columns striped across
// threads, 4 independent accumulators so 4 global loads are in flight.
// Table (120MB) is L2-resident (192MB L2), so loads are L2-bandwidth bound.
// ---------------------------------------------------------------------------
__global__ __launch_bounds__(320) void embed_mean_kernel(
    const int* __restrict__ x, const float* __restrict__ weight,
    float* __restrict__ y0) {
  __shared__ int toks[L_];
  __shared__ int firstZero;
  const int b = blockIdx.x;
  const int tid = threadIdx.x;
  if (tid == 0) firstZero = L_;
  __syncthreads();
  int localMin = L_;
  for (int l = tid; l < L_; l += 320) {
    int t = x[b * L_ + l];
    toks[l] = t;
    if (t == 0 && l < localMin) localMin = l;
  }
  if (localMin < L_) atomicMin(&firstZero, localMin);
  __syncthreads();
  // reference: len = L if last token nonzero else index of first zero
  const float len = (toks[L_ - 1] != 0) ? (float)L_ : (float)firstZero;
  const float inv = 1.0f / len;
  const int e = tid;
  if (e < E_) {
    float a0 = 0.f, a1 = 0.f, a2 = 0.f, a3 = 0.f;
    for (int l = 0; l < L_; l += 4) {
      a0 += weight[toks[l + 0] * E_ + e];
      a1 += weight[toks[l + 1] * E_ + e];
      a2 += weight[toks[l + 2] * E_ + e];
      a3 += weight[toks[l + 3] * E_ + e];
    }
    y0[b * E_ + e] = (a0 + a1 + a2 + a3) * inv;
  }
}

// ---------------------------------------------------------------------------
// Kernel B: z = relu(y0 @ w1^T + b1) via V_WMMA_F32_16X16X4_F32.
// 4 waves/block; each wave owns one 16x16 output tile, loops K=300 in steps
// of 4 (75 WMMAs). A and B frags are contiguous float2 per lane.
// ---------------------------------------------------------------------------
__global__ __launch_bounds__(128) void gemm1_relu_kernel(
    const float* __restrict__ y0, const float* __restrict__ w1,
    const float* __restrict__ b1, float* __restrict__ z) {
  const int lane = threadIdx.x & 31;
  const int wv = threadIdx.x >> 5;
  const int b0 = blockIdx.x * 16;                 // batch tile
  const int n0 = (blockIdx.y * 4 + wv) * 16;      // hidden tile
  const int r = lane & 15;
  const int kofs = (lane >> 4) << 1;              // 0 for lanes 0-15, 2 for 16-31
  const float* aptr = y0 + (b0 + r) * E_ + kofs;  // A[m][k], row-major
  const float* bptr = w1 + (n0 + r) * E_ + kofs;  // B[k][n] = w1[n][k]
  v8f c = {};
#pragma unroll 5
  for (int e0 = 0; e0 < E_; e0 += 4) {
    v2f a = *(const v2f*)(aptr + e0);
    v2f bb = *(const v2f*)(bptr + e0);
    // (neg_a, A, neg_b, B, c_mod, C, reuse_a, reuse_b)
    c = __builtin_amdgcn_wmma_f32_16x16x4_f32(false, a, false, bb, (short)0, c,
                                              false, false);
  }
  // C/D layout: VGPR v -> M = v + 8*(lane>=16), N = lane&15
  const int n = n0 + r;
  const float bias = b1[n];
  const int mbase = b0 + ((lane >> 4) << 3);
#pragma unroll
  for (int v = 0; v < 8; ++v) {
    float val = c[v] + bias;
    z[(mbase + v) * H_ + n] = val > 0.f ? val : 0.f;
  }
}

// ---------------------------------------------------------------------------
// Kernel C: out = z @ w2^T + b2 (O=5, tiny). One block per batch row,
// 64 threads strided over H, LDS reduction.
// ---------------------------------------------------------------------------
__global__ __launch_bounds__(64) void gemm2_kernel(
    const float* __restrict__ z, const float* __restrict__ w2,
    const float* __restrict__ b2, float* __restrict__ out) {
  __shared__ float red[O_][64];
  const int b = blockIdx.x;
  const int tid = threadIdx.x;
  float acc[O_] = {0.f, 0.f, 0.f, 0.f, 0.f};
  for (int h = tid; h < H_; h += 64) {
    const float zv = z[b * H_ + h];
#pragma unroll
    for (int o = 0; o < O_; ++o) acc[o] += zv * w2[o * H_ + h];
  }
#pragma unroll
  for (int o = 0; o < O_; ++o) red[o][tid] = acc[o];
  __syncthreads();
  if (tid < O_) {
    float s = 0.f;
#pragma unroll 8
    for (int i = 0; i < 64; ++i) s += red[tid][i];
    out[b * O_ + tid] = s + b2[tid];
  }
}

extern "C" void kernel_launch(void* const* d_in, const int* in_sizes, int n_in,
                              void* d_out, int out_size, void* d_ws,
                              size_t ws_size, hipStream_t stream) {
  (void)in_sizes; (void)n_in; (void)out_size; (void)ws_size;
  const int* x        = (const int*)d_in[0];     // [1024, 512] int32
  const float* weight = (const float*)d_in[1];   // [100000, 300]
  const float* w1     = (const float*)d_in[2];   // [512, 300]
  const float* b1     = (const float*)d_in[3];   // [512]
  const float* w2     = (const float*)d_in[4];   // [5, 512]
  const float* b2     = (const float*)d_in[5];   // [5]
  float* out = (float*)d_out;                    // [1024, 5]

  float* y0 = (float*)d_ws;                      // [1024, 300] mean embeddings
  float* z  = y0 + (size_t)B_ * E_;              // [1024, 512] hidden

  embed_mean_kernel<<<B_, 320, 0, stream>>>(x, weight, y0);
  gemm1_relu_kernel<<<dim3(B_ / 16, H_ / 64), 128, 0, stream>>>(y0, w1, b1, z);
  gemm2_kernel<<<B_, 64, 0, stream>>>(z, w2, b2, out);
}